// GCN_27350351741103
// MI455X (gfx1250) — compile-verified
//
#include <hip/hip_runtime.h>
#include <hip/hip_bf16.h>

typedef __attribute__((ext_vector_type(2))) float v2f;
typedef __attribute__((ext_vector_type(8))) float v8f;

#define NN 100000
#define NE 1600000
#define KIN 128
#define HD 64
#define NG 512

// ---------- degree / norm ----------
__global__ void k_deg_init(float* deg) {
    int i = blockIdx.x * 256 + threadIdx.x;
    if (i < NN) deg[i] = 1.0f;              // self-loop counts once
}

__global__ void k_deg_count(const int* __restrict__ dst, float* deg) {
    int e = blockIdx.x * 256 + threadIdx.x;
    if (e < NE) atomicAdd(&deg[dst[e]], 1.0f);
}

__global__ void k_rsqrt_(float* d) {
    int i = blockIdx.x * 256 + threadIdx.x;
    if (i < NN) d[i] = rsqrtf(d[i]);        // deg >= 1 always
}

// ---------- WMMA f32 GEMM: Out[N x 64] = A[N x K] * W[K x 64] ----------
// One wave per 16-row M tile; 4 accumulators cover all 64 output columns.
// A-frag (16x4 f32): lane M = lane&15; lanes 0-15 hold K {0,1}, lanes 16-31 hold K {2,3}.
// B-frag (4x16 f32): lane N = lane&15; same K split across lane halves.
// C/D (16x16 f32): VGPR v, lanes 0-15 -> M=v, lanes 16-31 -> M=v+8; N = lane&15.
//
// W is pre-packed into LDS as K-pair float2s: sW2[p*64+c] = {W[2p][c], W[2p+1][c]},
// so each B fragment is one 8-byte ds_load_b64 into an adjacent VGPR pair (no
// cross-register shuffles before the WMMA).
template<int K>
__global__ void __launch_bounds__(64) k_gemm_wmma(const float* __restrict__ A,
                                                  const float* __restrict__ W,
                                                  float* __restrict__ Out) {
    __shared__ v2f sW2[(K / 2) * HD];
    for (int j = threadIdx.x; j < (K / 2) * HD; j += 64) {
        int p = j >> 6, c = j & 63;
        v2f t = { W[(2 * p) * HD + c], W[(2 * p + 1) * HD + c] };
        sW2[j] = t;
    }
    __syncthreads();

    const int lane  = threadIdx.x & 31;
    const int wave  = threadIdx.x >> 5;
    const int mtile = blockIdx.x * 2 + wave;     // 6250 tiles total (100000/16)
    const int kh    = (lane >> 4) << 1;          // 0 or 2
    const int col   = lane & 15;                 // M for A-frag, N for B-frag

    const float* arow = A + (size_t)(mtile * 16 + col) * K;

    v8f acc0 = {}; v8f acc1 = {}; v8f acc2 = {}; v8f acc3 = {};
    for (int k = 0; k < K; k += 4) {
        v2f a = *(const v2f*)&arow[k + kh];                 // global_load_b64
        const v2f* wp = &sW2[((k + kh) >> 1) * HD + col];   // pair row p = (k+kh)/2
        v2f b0 = wp[0];                                     // ds_load_b64 each
        v2f b1 = wp[16];
        v2f b2 = wp[32];
        v2f b3 = wp[48];
        acc0 = __builtin_amdgcn_wmma_f32_16x16x4_f32(false, a, false, b0, (short)0, acc0, false, false);
        acc1 = __builtin_amdgcn_wmma_f32_16x16x4_f32(false, a, false, b1, (short)0, acc1, false, false);
        acc2 = __builtin_amdgcn_wmma_f32_16x16x4_f32(false, a, false, b2, (short)0, acc2, false, false);
        acc3 = __builtin_amdgcn_wmma_f32_16x16x4_f32(false, a, false, b3, (short)0, acc3, false, false);
    }

    const int mrow = mtile * 16 + ((lane >> 4) << 3);
    for (int v = 0; v < 8; ++v) {
        float* o = Out + (size_t)(mrow + v) * HD + col;
        o[0]  = acc0[v];
        o[16] = acc1[v];
        o[32] = acc2[v];
        o[48] = acc3[v];
    }
}

// ---------- accumulator init: bias + self-loop contribution ----------
__global__ void k_acc_init(const float* __restrict__ h, const float* __restrict__ dis,
                           const float* __restrict__ bias, float* __restrict__ acc) {
    int idx = blockIdx.x * 256 + threadIdx.x;
    if (idx < NN * HD) {
        int i = idx >> 6, c = idx & 63;
        float d = dis[i];
        acc[idx] = bias[c] + h[idx] * d * d;   // norm(self) = dis^2
    }
}

// ---------- edge scatter: acc[dst] += h[src] * dis[src]*dis[dst] ----------
__global__ void k_scatter(const int* __restrict__ src, const int* __restrict__ dst,
                          const float* __restrict__ dis, const float* __restrict__ h,
                          float* __restrict__ acc) {
    unsigned t = blockIdx.x * 256 + threadIdx.x;   // 16 lanes per edge
    unsigned e = t >> 4;
    if (e >= NE) return;
    int c = (t & 15) << 2;
    int s = src[e], d = dst[e];
    float nrm = dis[s] * dis[d];
    const float4 hv = *(const float4*)&h[(size_t)s * HD + c];
    float* ap = &acc[(size_t)d * HD + c];
    atomicAdd(ap + 0, hv.x * nrm);
    atomicAdd(ap + 1, hv.y * nrm);
    atomicAdd(ap + 2, hv.z * nrm);
    atomicAdd(ap + 3, hv.w * nrm);
}

__global__ void k_relu4(float4* x) {
    int i = blockIdx.x * 256 + threadIdx.x;        // NN*HD/4 elements
    if (i < NN * HD / 4) {
        float4 v = x[i];
        v.x = fmaxf(v.x, 0.0f);
        v.y = fmaxf(v.y, 0.0f);
        v.z = fmaxf(v.z, 0.0f);
        v.w = fmaxf(v.w, 0.0f);
        x[i] = v;
    }
}

// ---------- pooling ----------
__global__ void k_pool_zero(float* pooled, float* cnt) {
    int i = blockIdx.x * 256 + threadIdx.x;
    if (i < NG * HD) pooled[i] = 0.0f;
    if (i < NG) cnt[i] = 0.0f;
}

__global__ void k_pool(const float* __restrict__ h, const int* __restrict__ batch,
                       float* pooled, float* cnt) {
    int idx = blockIdx.x * 256 + threadIdx.x;
    if (idx < NN * HD) {
        int i = idx >> 6, c = idx & 63;
        int g = batch[i];
        atomicAdd(&pooled[g * HD + c], h[idx]);
        if (c == 0) atomicAdd(&cnt[g], 1.0f);
    }
}

__global__ void __launch_bounds__(64) k_final(const float* __restrict__ pooled,
                                              const float* __restrict__ cnt,
                                              const float* __restrict__ Wl,
                                              const float* __restrict__ bl,
                                              float* __restrict__ out) {
    __shared__ float red[64];
    int g = blockIdx.x, c = threadIdx.x;
    red[c] = pooled[g * HD + c] * Wl[c];
    __syncthreads();
    for (int s = 32; s > 0; s >>= 1) {
        if (c < s) red[c] += red[c + s];
        __syncthreads();
    }
    if (c == 0) out[g] = red[0] / fmaxf(cnt[g], 1.0f) + bl[0];
}

extern "C" void kernel_launch(void* const* d_in, const int* in_sizes, int n_in,
                              void* d_out, int out_size, void* d_ws, size_t ws_size,
                              hipStream_t stream) {
    const float* x    = (const float*)d_in[0];
    const int*   ei   = (const int*)d_in[1];
    const int*   srcp = ei;          // edge_index[0]
    const int*   dstp = ei + NE;     // edge_index[1]
    const int*   batch= (const int*)d_in[2];
    const float* W1   = (const float*)d_in[3];
    const float* b1   = (const float*)d_in[4];
    const float* W2   = (const float*)d_in[5];
    const float* b2   = (const float*)d_in[6];
    const float* Wl   = (const float*)d_in[7];
    const float* bl   = (const float*)d_in[8];
    float* out = (float*)d_out;

    char* ws = (char*)d_ws;
    float* dis    = (float*)(ws);                              // NN floats (deg -> dis in place)
    float* bufA   = (float*)(ws + 400000);                     // NN*HD floats
    float* bufB   = (float*)(ws + 400000 + 25600000);          // NN*HD floats
    float* pooled = (float*)(ws + 400000 + 51200000);          // NG*HD floats
    float* cnt    = (float*)(ws + 400000 + 51200000 + 131072); // NG floats

    dim3 b256(256);
    // normalization
    k_deg_init <<<(NN + 255) / 256, b256, 0, stream>>>(dis);
    k_deg_count<<<(NE + 255) / 256, b256, 0, stream>>>(dstp, dis);
    k_rsqrt_   <<<(NN + 255) / 256, b256, 0, stream>>>(dis);

    // layer 1
    k_gemm_wmma<KIN><<<NN / 32, dim3(64), 0, stream>>>(x, W1, bufA);
    k_acc_init<<<(NN * HD + 255) / 256, b256, 0, stream>>>(bufA, dis, b1, bufB);
    k_scatter <<<(NE * 16 + 255) / 256, b256, 0, stream>>>(srcp, dstp, dis, bufA, bufB);
    k_relu4   <<<(NN * HD / 4 + 255) / 256, b256, 0, stream>>>((float4*)bufB);

    // layer 2
    k_gemm_wmma<HD><<<NN / 32, dim3(64), 0, stream>>>(bufB, W2, bufA);
    k_acc_init<<<(NN * HD + 255) / 256, b256, 0, stream>>>(bufA, dis, b2, bufB);
    k_scatter <<<(NE * 16 + 255) / 256, b256, 0, stream>>>(srcp, dstp, dis, bufA, bufB);
    k_relu4   <<<(NN * HD / 4 + 255) / 256, b256, 0, stream>>>((float4*)bufB);

    // pooling + head
    k_pool_zero<<<(NG * HD + 255) / 256, b256, 0, stream>>>(pooled, cnt);
    k_pool     <<<(NN * HD + 255) / 256, b256, 0, stream>>>(bufB, batch, pooled, cnt);
    k_final    <<<NG, dim3(64), 0, stream>>>(pooled, cnt, Wl, bl, out);
}